// splatter_78563541778948
// MI455X (gfx1250) — compile-verified
//
#include <hip/hip_runtime.h>
#include <hip/hip_bf16.h>

// CDNA5 / gfx1250 5x5 fp32 "SAME" convolution via V_WMMA_F32_16X16X4_F32.
//
// out[y,x] = sum_{a,b in [0,5)} in[y+2-a, x+2-b] * kernel[a,b]   (true conv)
//
// Per 16x16 output tile:  Out = sum_{dy=0..4} A_dy(16x20) x B_dy(20x16),
// banded B_dy[j,c] = wflip[dy][j-c], split into 5 K=4 chunks
// -> 25 v_wmma_f32_16x16x4_f32 per tile, dual accumulators.
//
// One wave processes a strip of 8 tiles along x so the 25 B fragments
// (tile-independent, ~250 VALU to build) and weight loads are amortized.

typedef __attribute__((ext_vector_type(2))) float v2f;
typedef __attribute__((ext_vector_type(8))) float v8f;

#define IMG_H 4096
#define IMG_W 4096
#define TILES_X 256          // 4096/16
#define TILES_Y 256
#define STRIP   8            // tiles per wave along x
#define STRIPS_X (TILES_X / STRIP)   // 32

__device__ __forceinline__ int clampi(int v, int lo, int hi) {
    return v < lo ? lo : (v > hi ? hi : v);
}

// Clamped-address load + select-to-zero (value select; address always valid).
__device__ __forceinline__ float ld_guard(const float* __restrict__ p, int r, int c) {
    int rc = clampi(r, 0, IMG_H - 1);
    int cc = clampi(c, 0, IMG_W - 1);
    float v = p[rc * IMG_W + cc];
    return (r == rc && c == cc) ? v : 0.0f;
}

__global__ __launch_bounds__(256) void conv5x5_wmma_f32(
        const float* __restrict__ in,
        const float* __restrict__ kern,
        float* __restrict__ out) {
    const int lane = threadIdx.x & 31;
    const int wave = threadIdx.x >> 5;                   // 0..7
    const int strip = blockIdx.x * 8 + wave;             // 0..8191 (wave-uniform)
    const int ty  = strip / STRIPS_X;                    // 0..255
    const int tx0 = (strip % STRIPS_X) * STRIP;          // first tile in strip
    const int y0  = ty << 4;

    const int m  = lane & 15;   // M row (A) / N col (B,C)
    const int hi = lane >> 4;   // 0: K={0,1}, 1: K={2,3} in A/B fragments

    // Flipped weights: wflip[dy*5+dx] = kernel[4-dy][4-dx] = kern[24 - (dy*5+dx)]
    float wc[25];
#pragma unroll
    for (int i = 0; i < 25; ++i) wc[i] = kern[24 - i];

    // 25 B fragments, built once per wave, reused for all 8 tiles.
    // B layout (4x16, 2 VGPRs): VGPR0 = {K=0 lanes0-15 | K=2 lanes16-31},
    //                           VGPR1 = {K=1           | K=3            }.
    // B[j, c] = wc[dy][j-c] for 0 <= j-c <= 4 else 0;  j = kc*4 + hi*2 + {0,1}, c = m.
    v2f bfrag[25];
#pragma unroll
    for (int dy = 0; dy < 5; ++dy) {
#pragma unroll
        for (int kc = 0; kc < 5; ++kc) {
            const int j0 = kc * 4 + hi * 2;
            const int d0 = j0 - m;
            const int d1 = d0 + 1;
            float b0 = 0.0f, b1 = 0.0f;
#pragma unroll
            for (int t = 0; t < 5; ++t) {
                const float w = wc[dy * 5 + t];
                b0 = (d0 == t) ? w : b0;
                b1 = (d1 == t) ? w : b1;
            }
            bfrag[dy * 5 + kc] = (v2f){b0, b1};
        }
    }

    const int rowInterior = (ty > 0) & (ty < TILES_Y - 1);
    // Row base for the unguarded path (strip-invariant part).
    const float* rowbase = in + (y0 + m - 2) * IMG_W + hi * 2;

    for (int t = 0; t < STRIP; ++t) {
        const int tx = tx0 + t;
        const int x0 = tx << 4;

        // ---- Stage all 25 A fragments into registers (branch only here). ----
        // A[m,k] = in[y0+m+dy-2, x0-2 + kc*4 + hi*2 + k],  k = {0,1} per lane half.
        v2f afrag[25];
        const int interior = rowInterior & (tx > 0) & (tx < TILES_X - 1);
        if (__builtin_amdgcn_readfirstlane(interior)) {
            const float* p0 = rowbase + (x0 - 2);
#pragma unroll
            for (int dy = 0; dy < 5; ++dy) {
#pragma unroll
                for (int kc = 0; kc < 5; ++kc) {
                    v2f a;
                    __builtin_memcpy(&a, p0 + dy * IMG_W + kc * 4, 8); // 4B-aligned 8B load
                    afrag[dy * 5 + kc] = a;
                }
            }
        } else {
            const int r0  = y0 + m - 2;
            const int c00 = x0 - 2 + hi * 2;
#pragma unroll
            for (int dy = 0; dy < 5; ++dy) {
#pragma unroll
                for (int kc = 0; kc < 5; ++kc) {
                    const int cb = c00 + kc * 4;
                    v2f a;
                    a.x = ld_guard(in, r0 + dy, cb);
                    a.y = ld_guard(in, r0 + dy, cb + 1);
                    afrag[dy * 5 + kc] = a;
                }
            }
        }

        // ---- 25 WMMAs, outside any branch (EXEC all-ones), dual accumulators. ----
        v8f acc0 = {}, acc1 = {};
#pragma unroll
        for (int i = 0; i < 25; ++i) {
            if (i & 1)
                acc1 = __builtin_amdgcn_wmma_f32_16x16x4_f32(
                    false, afrag[i], false, bfrag[i], (short)0, acc1, false, false);
            else
                acc0 = __builtin_amdgcn_wmma_f32_16x16x4_f32(
                    false, afrag[i], false, bfrag[i], (short)0, acc0, false, false);
        }

        // ---- Store. C/D layout: VGPR v -> row y0 + v + 8*hi, col x0 + m. ----
        float* orow = out + (y0 + hi * 8) * IMG_W + x0 + m;
#pragma unroll
        for (int v = 0; v < 8; ++v) {
            orow[v * IMG_W] = acc0[v] + acc1[v];
        }
    }
}

extern "C" void kernel_launch(void* const* d_in, const int* in_sizes, int n_in,
                              void* d_out, int out_size, void* d_ws, size_t ws_size,
                              hipStream_t stream) {
    const float* in   = (const float*)d_in[0];   // 4096*4096 f32
    const float* kern = (const float*)d_in[1];   // 5*5 f32
    float* out = (float*)d_out;                  // 4096*4096 f32

    const int strips = TILES_Y * STRIPS_X;       // 8192 waves, 8 waves/block
    dim3 grid(strips / 8), block(256);
    conv5x5_wmma_f32<<<grid, block, 0, stream>>>(in, kern, out);
}